// GraphConvolution_558345749111
// MI455X (gfx1250) — compile-verified
//
#include <hip/hip_runtime.h>
#include <math.h>

typedef __attribute__((ext_vector_type(2))) float v2f;
typedef __attribute__((ext_vector_type(4))) float v4f;
typedef __attribute__((ext_vector_type(8))) float v8f;

#define N_NODES 100000
#define N_EDGES 1600000
#define IN_F    256
#define OUT_F   128

#define ROWS_PER_BLK 32          // 2 x 16-row WMMA tiles; 100000/32 = 3125 exact

// ---------------------------------------------------------------------------
// Kernel 1: support = X @ W via V_WMMA_F32_16X16X4_F32
// grid = N_NODES/32 blocks, 256 threads (8 waves). Wave w owns columns
// [16w, 16w+16) and TWO 16-row tiles, sharing each B fragment across both
// (halves W L2 traffic to ~400 MB total and doubles WMMA density per load).
// X staging loads are non-temporal: X is 102 MB read-once and must not evict
// support/W from the 192 MB L2.
// ---------------------------------------------------------------------------
__global__ __launch_bounds__(256) void gcn_gemm_wmma(
    const float* __restrict__ X,        // [N_NODES, IN_F]
    const float* __restrict__ W,        // [IN_F, OUT_F]
    float* __restrict__ support)        // [N_NODES, OUT_F]
{
    __shared__ float ldsX[ROWS_PER_BLK * IN_F];   // 32 KB tile of X

    const int t       = threadIdx.x;
    const int rowBase = blockIdx.x * ROWS_PER_BLK;

    // Stage 32x256 fp32 X tile into LDS, coalesced b128, TH=NT on global.
    {
        const v4f* src = (const v4f*)(X + (size_t)rowBase * IN_F);
        v4f*       dst = (v4f*)ldsX;
#pragma unroll
        for (int i = 0; i < 8; ++i) {             // 8 * 256 v4f = 8192 floats
            const int idx = i * 256 + t;
            dst[idx] = __builtin_nontemporal_load(&src[idx]);
        }
    }
    __syncthreads();

    const int wave   = t >> 5;
    const int lane   = t & 31;
    const int half   = lane >> 4;   // 0: lanes 0-15, 1: lanes 16-31
    const int lane16 = lane & 15;
    const int n0     = wave * 16;   // column tile base
    const int koff   = half * 2;    // A/B K sub-offset per ISA fp32 layout

    // A-matrix 16x4 fp32 layout: lanes 0-15 row M=lane (K=0 in v0, K=1 in v1),
    // lanes 16-31 row M=lane-16 (K=2 in v0, K=3 in v1).
    const float* ldsA0 = ldsX + lane16 * IN_F;            // rows  0..15
    const float* ldsA1 = ldsX + (16 + lane16) * IN_F;     // rows 16..31

    v8f acc0 = {};
    v8f acc1 = {};
#pragma unroll 4
    for (int k0 = 0; k0 < IN_F; k0 += 4) {
        v2f b;
        // B-matrix 4x16 fp32: row K striped across lanes within a VGPR,
        // lanes 0-15 hold K=k0+0/k0+1, lanes 16-31 hold K=k0+2/k0+3.
        b.x = W[(size_t)(k0 + koff)     * OUT_F + n0 + lane16];
        b.y = W[(size_t)(k0 + koff + 1) * OUT_F + n0 + lane16];

        v2f a0 = *(const v2f*)(ldsA0 + k0 + koff);        // ds_load_b64
        v2f a1 = *(const v2f*)(ldsA1 + k0 + koff);

        acc0 = __builtin_amdgcn_wmma_f32_16x16x4_f32(
                   false, a0, false, b, (short)0, acc0, false, false);
        acc1 = __builtin_amdgcn_wmma_f32_16x16x4_f32(
                   false, a1, false, b, (short)0, acc1, false, false);
    }

    // C/D 16x16 f32 layout: VGPR r -> (row r, col lane) lanes 0-15,
    //                                  (row r+8, col lane-16) lanes 16-31.
    float* out0 = support + (size_t)rowBase * OUT_F + n0 + lane16;
    float* out1 = out0 + (size_t)16 * OUT_F;
#pragma unroll
    for (int r = 0; r < 8; ++r) {
        out0[(size_t)(r + half * 8) * OUT_F] = acc0[r];
        out1[(size_t)(r + half * 8) * OUT_F] = acc1[r];
    }
}

// ---------------------------------------------------------------------------
// Kernel 2: out[node] = tanh(sum_{e: dst==node} val[e]*support[src[e]] + bias)
// One wave32 per node. edge_dst is sorted -> per-node edge range by binary
// search (wave-uniform), no atomics, each output row written exactly once.
// support (51.2 MB) is L2-resident (192 MB L2); out stores are non-temporal
// so the 51 MB write-once stream doesn't evict it.
// ---------------------------------------------------------------------------
__global__ __launch_bounds__(256) void gcn_aggregate(
    const float* __restrict__ support,   // [N_NODES, OUT_F]
    const int*   __restrict__ edge_src,  // [N_EDGES]
    const int*   __restrict__ edge_dst,  // [N_EDGES] sorted
    const float* __restrict__ edge_val,  // [N_EDGES]
    const float* __restrict__ bias,      // [OUT_F]
    float* __restrict__ out)             // [N_NODES, OUT_F]
{
    const int wave = threadIdx.x >> 5;
    const int lane = threadIdx.x & 31;
    const int node = blockIdx.x * 8 + wave;
    if (node >= N_NODES) return;

    // lower_bound(edge_dst, node)
    int lo = 0, hi = N_EDGES;
    while (lo < hi) {
        const int mid = (lo + hi) >> 1;
        if (edge_dst[mid] < node) lo = mid + 1; else hi = mid;
    }
    const int e0 = lo;
    // upper_bound(edge_dst, node)
    hi = N_EDGES;
    while (lo < hi) {
        const int mid = (lo + hi) >> 1;
        if (edge_dst[mid] <= node) lo = mid + 1; else hi = mid;
    }
    const int e1 = lo;

    const int f0 = lane * 4;                  // 32 lanes * 4 floats = 128 feats
    v4f acc = {0.f, 0.f, 0.f, 0.f};

#pragma unroll 4
    for (int e = e0; e < e1; ++e) {
        const int   src = edge_src[e];
        const float val = edge_val[e];
        const v4f   sv  = *(const v4f*)(support + (size_t)src * OUT_F + f0);
        acc += val * sv;                      // ext-vector FMA per component
    }

    const v4f b = *(const v4f*)(bias + f0);
    v4f r;
    r.x = tanhf(acc.x + b.x);
    r.y = tanhf(acc.y + b.y);
    r.z = tanhf(acc.z + b.z);
    r.w = tanhf(acc.w + b.w);
    __builtin_nontemporal_store(r, (v4f*)(out + (size_t)node * OUT_F + f0));
}

// ---------------------------------------------------------------------------
extern "C" void kernel_launch(void* const* d_in, const int* in_sizes, int n_in,
                              void* d_out, int out_size, void* d_ws, size_t ws_size,
                              hipStream_t stream) {
    const float* x    = (const float*)d_in[0];   // [N_NODES, IN_F]
    const float* w    = (const float*)d_in[1];   // [IN_F, OUT_F]
    const float* bias = (const float*)d_in[2];   // [OUT_F]
    const int*   esrc = (const int*)  d_in[3];   // [N_EDGES]
    const int*   edst = (const int*)  d_in[4];   // [N_EDGES] sorted
    const float* eval = (const float*)d_in[5];   // [N_EDGES]

    float* out     = (float*)d_out;              // [N_NODES, OUT_F]
    float* support = (float*)d_ws;               // 51.2 MB scratch

    gcn_gemm_wmma<<<N_NODES / ROWS_PER_BLK, 256, 0, stream>>>(x, w, support);
    gcn_aggregate<<<(N_NODES + 7) / 8, 256, 0, stream>>>(support, esrc, edst,
                                                         eval, bias, out);
}